// Shift_11261404250938
// MI455X (gfx1250) — compile-verified
//
#include <hip/hip_runtime.h>
#include <stdint.h>

// Problem constants (match reference): B=256, C=3, H=224, W=224
#define CCH   3
#define HH    224
#define WW    224
#define HW    (HH * WW)
#define CHW   (CCH * HW)
#define THREADS 512
#define NWAVES  (THREADS / 32)   // 16 wave32s per block

// out = (shifted - min(shifted)) / (max(shifted) - min(shifted))
// The reference's standardize step is a positive-scale affine map, so it
// cancels exactly against the subsequent min-max normalization.
__global__ __launch_bounds__(THREADS, 1)
void shift_minmax_norm(const float* __restrict__ x,
                       const float* __restrict__ sfy,
                       const float* __restrict__ sfx,
                       float* __restrict__ out)
{
    const int b = blockIdx.x;
    const float* img = x   + (size_t)b * CHW;
    float*       o   = out + (size_t)b * CHW;

    const int lane = threadIdx.x & 31;
    const int wave = threadIdx.x >> 5;

    // int() truncates toward zero, same as jnp.trunc -> int32
    const int sy = (int)((sfy[b] * 2.0f - 1.0f) * 0.1f * (float)HH);
    const int sx = (int)((sfx[b] * 2.0f - 1.0f) * 0.1f * (float)WW);

    // Source rectangle that survives the shift (everything else becomes 0)
    const int iy0 = sy > 0 ? sy : 0;
    const int iy1 = sy < 0 ? (HH + sy) : HH;
    const int jx0 = sx > 0 ? sx : 0;
    const int jx1 = sx < 0 ? (WW + sx) : WW;
    const bool hasZero = (sy != 0) || (sx != 0);

    float mn = hasZero ? 0.0f :  3.0e38f;
    float mx = hasZero ? 0.0f : -3.0e38f;

    // ---------- Pass 1: min/max over the valid source rectangle ----------
    // HBM-bound pass: vectorized b128 row reads (rows are 16B aligned:
    // W*4 = 896 = 56*16), scalar head/tail for the shift misalignment.
    const int a0 = (jx0 + 3) & ~3;   // first 4-aligned index >= jx0
    const int a1 = jx1 & ~3;         // last 4-aligned end   <= jx1
    const int q0 = a0 >> 2;
    const int q1 = a1 >> 2;

    for (int c = 0; c < CCH; ++c) {
        const float* chan = img + c * HW;
        for (int ii = iy0 + wave; ii < iy1; ii += NWAVES) {
            const float*  row  = chan + ii * WW;
            const float4* row4 = (const float4*)row;
            // head (<=3 elems) and tail (<=3 elems), lane-predicated
            {
                int hj = jx0 + lane;
                if (hj < a0) { float v = row[hj]; mn = fminf(mn, v); mx = fmaxf(mx, v); }
                int tj = a1 + lane;
                if (tj < jx1) { float v = row[tj]; mn = fminf(mn, v); mx = fmaxf(mx, v); }
            }
            // aligned main body: global_load_b128 per lane
            for (int q = q0 + lane; q < q1; q += 32) {
                float4 v = row4[q];
                mn = fminf(mn, fminf(fminf(v.x, v.y), fminf(v.z, v.w)));
                mx = fmaxf(mx, fmaxf(fmaxf(v.x, v.y), fmaxf(v.z, v.w)));
            }
        }
    }
    // wave32 butterfly reduction
    for (int off = 16; off > 0; off >>= 1) {
        mn = fminf(mn, __shfl_xor(mn, off, 32));
        mx = fmaxf(mx, __shfl_xor(mx, off, 32));
    }
    __shared__ float smn[NWAVES], smx[NWAVES];
    __shared__ float bcast[2];
    if (lane == 0) { smn[wave] = mn; smx[wave] = mx; }
    __syncthreads();
    if (threadIdx.x == 0) {
        float a = smn[0], z = smx[0];
        for (int wv = 1; wv < NWAVES; ++wv) {
            a = fminf(a, smn[wv]);
            z = fmaxf(z, smx[wv]);
        }
        bcast[0] = a;
        bcast[1] = 1.0f / (z - a);
    }
    __syncthreads();
    const float gmn  = bcast[0];
    const float ginv = bcast[1];

    // ---------- Pass 2: shifted gather -> normalize -> NT store ----------
    // Reads hit L2 (input resident from pass 1; NT output stores keep it
    // there). Source rows are staged into LDS with the CDNA5 async
    // global->LDS engine (ASYNCcnt), double-buffered per wave: issue chunk
    // k+1 while consuming chunk k. Per-lane b32 grain is the only
    // alignment-safe choice (column shift sx => 4B alignment only).
    __shared__ float stage[NWAVES][2][32];

    for (int c = 0; c < CCH; ++c) {
        for (int i = wave; i < HH; i += NWAVES) {   // 224 % 16 == 0: uniform trips
            const int  ip    = i + sy;
            const bool rowOK = (ip >= 0) && (ip < HH);

#define ISSUE_CHUNK(K)                                                         \
            do {                                                               \
                int jp_   = 32 * (K) + lane + sx;                              \
                int sidx_ = c * HW + ip * WW + jp_;                            \
                sidx_ = sidx_ < 0 ? 0 : (sidx_ >= CHW ? (CHW - 1) : sidx_);    \
                uint64_t gaddr_ = (uint64_t)(uintptr_t)(img + sidx_);          \
                uint32_t laddr_ =                                              \
                    (uint32_t)(uintptr_t)(&stage[wave][(K) & 1][lane]);        \
                asm volatile("global_load_async_to_lds_b32 %0, %1, off"        \
                             :: "v"(laddr_), "v"(gaddr_) : "memory");          \
            } while (0)

            ISSUE_CHUNK(0);
            for (int k = 0; k < 7; ++k) {            // 224 = 7 chunks of 32
                if (k < 6) {
                    ISSUE_CHUNK(k + 1);
                    asm volatile("s_wait_asynccnt 1" ::: "memory");
                } else {
                    asm volatile("s_wait_asynccnt 0" ::: "memory");
                }
                const int  j  = 32 * k + lane;
                const int  jp = j + sx;
                const bool ok = rowOK && (jp >= 0) && (jp < WW);
                float v = ok ? stage[wave][k & 1][lane] : 0.0f;
                // NT store: don't pollute L2 with the output stream
                __builtin_nontemporal_store((v - gmn) * ginv,
                                            &o[c * HW + i * WW + j]);
            }
#undef ISSUE_CHUNK
        }
    }
}

extern "C" void kernel_launch(void* const* d_in, const int* in_sizes, int n_in,
                              void* d_out, int out_size, void* d_ws, size_t ws_size,
                              hipStream_t stream) {
    (void)n_in; (void)out_size; (void)d_ws; (void)ws_size;
    const float* x  = (const float*)d_in[0];
    const float* fy = (const float*)d_in[1];
    const float* fx = (const float*)d_in[2];
    float* out = (float*)d_out;

    const int batch = in_sizes[1];   // B = 256 per reference
    hipLaunchKernelGGL(shift_minmax_norm, dim3(batch), dim3(THREADS), 0, stream,
                       x, fy, fx, out);
}